// Encoder_67688684585543
// MI455X (gfx1250) — compile-verified
//
#include <hip/hip_runtime.h>

typedef __attribute__((ext_vector_type(2))) float v2f;
typedef __attribute__((ext_vector_type(8))) float v8f;

#define BB 8
#define KK 21
#define KPAD 32
#define HW 4096
#define FF 2048
#define EE 256

// ---------------------------------------------------------------------------
// Kernel 0: zero the per-class pixel counters.
// ---------------------------------------------------------------------------
__global__ void k_init_counts(int* __restrict__ counts) {
    int t = threadIdx.x;
    if (t < BB * KPAD) counts[t] = 0;
}

// ---------------------------------------------------------------------------
// Kernel 1: per-pixel argmax over K classes + per-class counts.
// outputs layout: [B, K, H*W]; coalesced per-k reads (stride HW between k).
// ---------------------------------------------------------------------------
__global__ void k_argmax(const float* __restrict__ outputs,
                         int* __restrict__ cls,
                         int* __restrict__ counts) {
    int idx = blockIdx.x * blockDim.x + threadIdx.x;   // 0 .. B*HW-1
    if (idx >= BB * HW) return;
    int b = idx / HW;
    int p = idx - b * HW;
    const float* o = outputs + (size_t)b * KK * HW + p;
    float best = o[0];
    int bi = 0;
#pragma unroll
    for (int k = 1; k < KK; ++k) {
        float v = o[(size_t)k * HW];
        if (v > best) { best = v; bi = k; }   // strict > : first-max, matches jnp.argmax
    }
    cls[idx] = bi;
    atomicAdd(&counts[b * KPAD + bi], 1);
}

// ---------------------------------------------------------------------------
// Kernel 2: segmented reduction as WMMA vs one-hot, fp32-exact.
// One 128-thread block (4 waves) per (b, 16-f tile). Each wave reduces a
// 1024-pixel quarter (=> 4096 resident waves for latency hiding at HBM rate),
// then a deterministic LDS tree-combine folds the four partial C tiles.
//   A (16x4 f32): lane l<16 holds feats[f0+l][p+0..1]; lanes>=16 p+2..3.
//   B (4x16 f32): one-hot rows built from cls via v_cmp/v_cndmask.
// Two independent accumulator pairs per wave break the WMMA C-dependency chain.
// Writes feat_set[b][f][KPAD] already divided by (count + 0.01).
// ---------------------------------------------------------------------------
__global__ void k_reduce_wmma(const float* __restrict__ feats,
                              const int* __restrict__ cls,
                              const int* __restrict__ counts,
                              float* __restrict__ fs) {
    int bt = blockIdx.x;                 // 0 .. B*(F/16)-1
    int b  = bt / (FF / 16);
    int ft = bt - b * (FF / 16);
    int f0 = ft * 16;

    int tid  = threadIdx.x;              // 0..127
    int wave = tid >> 5;                 // 0..3  : pixel-range quarter
    int lane = tid & 31;
    int half = lane >> 4;                // 0: k-rows 0,1   1: k-rows 2,3
    int l    = lane & 15;
    int p0   = wave * (HW / 4);          // 1024 pixels per wave

    const float* fbase = feats + (size_t)b * FF * HW + (size_t)(f0 + l) * HW + 2 * half + p0;
    const int*   cbase = cls + b * HW + 2 * half + p0;

    v8f c0a = {}, c1a = {}, c0b = {}, c1b = {};

    for (int p = 0; p < HW / 4; p += 8) {
        v2f a0 = *(const v2f*)(fbase + p);
        v2f a1 = *(const v2f*)(fbase + p + 4);
        int2 cc0 = *(const int2*)(cbase + p);
        int2 cc1 = *(const int2*)(cbase + p + 4);

        v2f b00, b01, b10, b11;
        b00.x = (cc0.x == l)      ? 1.0f : 0.0f;
        b00.y = (cc0.y == l)      ? 1.0f : 0.0f;
        b01.x = (cc0.x == l + 16) ? 1.0f : 0.0f;
        b01.y = (cc0.y == l + 16) ? 1.0f : 0.0f;
        b10.x = (cc1.x == l)      ? 1.0f : 0.0f;
        b10.y = (cc1.y == l)      ? 1.0f : 0.0f;
        b11.x = (cc1.x == l + 16) ? 1.0f : 0.0f;
        b11.y = (cc1.y == l + 16) ? 1.0f : 0.0f;

        c0a = __builtin_amdgcn_wmma_f32_16x16x4_f32(false, a0, false, b00, (short)0, c0a, false, false);
        c1a = __builtin_amdgcn_wmma_f32_16x16x4_f32(false, a0, false, b01, (short)0, c1a, false, false);
        c0b = __builtin_amdgcn_wmma_f32_16x16x4_f32(false, a1, false, b10, (short)0, c0b, false, false);
        c1b = __builtin_amdgcn_wmma_f32_16x16x4_f32(false, a1, false, b11, (short)0, c1b, false, false);
    }

    // --- deterministic cross-wave combine through LDS (stride 17: conflict-free) ---
    __shared__ float lds[4 * 32 * 17];
    float* myl = &lds[(wave * 32 + lane) * 17];
#pragma unroll
    for (int r = 0; r < 8; ++r) {
        myl[r]     = c0a[r] + c0b[r];    // classes 0..15
        myl[8 + r] = c1a[r] + c1b[r];    // classes 16..31
    }
    __syncthreads();

    if (wave == 0) {
        float inv0 = 1.0f / ((float)counts[b * KPAD + l]      + 0.01f);
        float inv1 = 1.0f / ((float)counts[b * KPAD + l + 16] + 0.01f);
        float* out0 = fs + ((size_t)b * FF + f0) * KPAD;
#pragma unroll
        for (int r = 0; r < 16; ++r) {
            float s = lds[(0 * 32 + lane) * 17 + r]
                    + lds[(1 * 32 + lane) * 17 + r]
                    + lds[(2 * 32 + lane) * 17 + r]
                    + lds[(3 * 32 + lane) * 17 + r];
            int m = (r & 7) + 8 * half;                    // f within tile
            if (r < 8) out0[(size_t)m * KPAD + l]      = s * inv0;
            else       out0[(size_t)m * KPAD + l + 16] = s * inv1;
        }
    }
}

// ---------------------------------------------------------------------------
// Kernel 3: out[b,e,k] = sum_f w[e,f] * fs[b,f,k] + bias[e], via fp32 WMMA.
// One wave per (b, e-tile of 16, k-tile of 16). 512 k-steps over F.
// Everything (w: 2 MB, fs: 2 MB) is L2-resident; ~64 MB L2 traffic total.
// ---------------------------------------------------------------------------
__global__ void k_gemm_wmma(const float* __restrict__ w,
                            const float* __restrict__ bias,
                            const float* __restrict__ fs,
                            float* __restrict__ out) {
    int t = blockIdx.x;                  // 0 .. B*16*2-1
    int b   = t / 32;
    int rem = t - b * 32;
    int mt  = rem >> 1;                  // 0..15 (E tiles)
    int nt  = rem & 1;                   // 0..1  (K tiles)
    int e0 = mt * 16, k0 = nt * 16;

    int lane = threadIdx.x;
    int half = lane >> 4;
    int l    = lane & 15;

    const float* abase = w + (size_t)(e0 + l) * FF + 2 * half;          // A rows = e
    const float* bbase = fs + (size_t)b * FF * KPAD
                            + (size_t)(2 * half) * KPAD + k0 + l;       // B rows = f

    v8f acc0 = {}, acc1 = {};
    for (int f = 0; f < FF; f += 8) {
        v2f a0 = *(const v2f*)(abase + f);
        v2f a1 = *(const v2f*)(abase + f + 4);
        v2f bv0, bv1;
        bv0.x = bbase[(size_t)f * KPAD];
        bv0.y = bbase[(size_t)(f + 1) * KPAD];
        bv1.x = bbase[(size_t)(f + 4) * KPAD];
        bv1.y = bbase[(size_t)(f + 5) * KPAD];
        acc0 = __builtin_amdgcn_wmma_f32_16x16x4_f32(false, a0, false, bv0, (short)0, acc0, false, false);
        acc1 = __builtin_amdgcn_wmma_f32_16x16x4_f32(false, a1, false, bv1, (short)0, acc1, false, false);
    }

    int kk = k0 + l;
    if (kk < KK) {
#pragma unroll
        for (int r = 0; r < 8; ++r) {
            int e = e0 + r + 8 * half;
            out[((size_t)b * EE + e) * KK + kk] = (acc0[r] + acc1[r]) + bias[e];
        }
    }
}

// ---------------------------------------------------------------------------
extern "C" void kernel_launch(void* const* d_in, const int* in_sizes, int n_in,
                              void* d_out, int out_size, void* d_ws, size_t ws_size,
                              hipStream_t stream) {
    const float* outputs = (const float*)d_in[0];   // [8,21,64,64]
    const float* feats   = (const float*)d_in[1];   // [8,2048,64,64]
    const float* w_proj  = (const float*)d_in[2];   // [256,2048]
    const float* b_proj  = (const float*)d_in[3];   // [256]
    float* out = (float*)d_out;                     // [8,256,21]

    char* ws = (char*)d_ws;
    int*   cls    = (int*)ws;                                     // B*HW ints   (128 KB)
    int*   counts = (int*)(ws + (size_t)BB * HW * 4);             // B*KPAD ints (1 KB)
    float* fs     = (float*)(ws + (size_t)BB * HW * 4 + BB * KPAD * 4); // B*F*KPAD f32 (2 MB)

    k_init_counts<<<1, 256, 0, stream>>>(counts);
    k_argmax<<<(BB * HW + 255) / 256, 256, 0, stream>>>(outputs, cls, counts);
    k_reduce_wmma<<<BB * (FF / 16), 128, 0, stream>>>(feats, cls, counts, fs);
    k_gemm_wmma<<<BB * 16 * 2, 32, 0, stream>>>(w_proj, b_proj, fs, out);
}